// GCN_61048665145867
// MI455X (gfx1250) — compile-verified
//
#include <hip/hip_runtime.h>

typedef __attribute__((ext_vector_type(2))) float v2f;
typedef __attribute__((ext_vector_type(8))) float v8f;

#define NFEAT 128   // K dimension of both GEMMs (N_FEAT == HID == 128)

// ---------------------------------------------------------------------------
// deg[i] = 1.0 (self loop)
__global__ void k_init_deg(float* deg, int n) {
    int i = blockIdx.x * blockDim.x + threadIdx.x;
    if (i < n) deg[i] = 1.0f;
}

// deg[dst[e]] += 1
__global__ void k_deg_accum(const int* __restrict__ dst, float* deg, int E) {
    int e = blockIdx.x * blockDim.x + threadIdx.x;
    if (e < E) atomicAdd(&deg[dst[e]], 1.0f);
}

// dis[i] = rsqrt(deg[i])
__global__ void k_rsqrt(float* d, int n) {
    int i = blockIdx.x * blockDim.x + threadIdx.x;
    if (i < n) d[i] = rsqrtf(d[i]);
}

// ---------------------------------------------------------------------------
// Fused GCNConv matmul + self-loop epilogue:
//   XW    [M x ncols] = (RELU ? relu(A) : A)[M x 128] @ B[128 x ncols]
//   Cinit [M x ncols] = XW * dis[row]^2 + bias[col]      (aggregation init)
//
// One wave computes a 16 x (16*NT) strip with NT fp32-WMMA accumulators,
// sharing one A fragment across the NT tiles each k-step.
//
// FP32 operand layouts (ISA 7.12.2):
//   A 16x4:  lanes 0-15: M=lane,    v0=K0, v1=K1 ; lanes 16-31: M=lane-16, v0=K2, v1=K3
//   B 4x16:  mirrored with N in place of M
//   C/D 16x16: vr -> M = r + 8*(lane>=16), N = lane&15
//
// GUARD=false requires gridDim.y * 16*NT == ncols exactly.
template<int NT, bool RELU, bool GUARD>
__global__ void k_wmma_gemm_fused(const float* __restrict__ A,
                                  const float* __restrict__ B,
                                  const float* __restrict__ dis,
                                  const float* __restrict__ bias,
                                  float* __restrict__ XW,
                                  float* __restrict__ Cinit,
                                  int nrows, int ncols) {
    const int lane = threadIdx.x & 31;
    const int half = lane >> 4;      // 0: lanes 0-15, 1: lanes 16-31
    const int l    = lane & 15;

    int arow = blockIdx.x * 16 + l;            // A row this lane loads
    if (arow >= nrows) arow = nrows - 1;       // clamp: EXEC must stay full for WMMA
    const float* __restrict__ Arow = A + (size_t)arow * NFEAT;

    // Per-tile column (clamped) + multiplicative mask: branch-free OOB handling.
    int   col [NT];
    float mask[NT];
    #pragma unroll
    for (int t = 0; t < NT; ++t) {
        const int c0 = blockIdx.y * (16 * NT) + t * 16 + l;
        if (GUARD) {
            const bool ok = (c0 < ncols);
            col [t] = ok ? c0 : (ncols - 1);
            mask[t] = ok ? 1.0f : 0.0f;
        } else {
            col [t] = c0;
            mask[t] = 1.0f;
        }
    }

    v8f acc[NT];
    #pragma unroll
    for (int t = 0; t < NT; ++t) acc[t] = (v8f){0, 0, 0, 0, 0, 0, 0, 0};

    #pragma unroll 4
    for (int k0 = 0; k0 < NFEAT; k0 += 4) {
        const int ka = k0 + 2 * half;          // this half-wave covers K = ka, ka+1
        v2f a;
        a.x = Arow[ka];
        a.y = Arow[ka + 1];
        if (RELU) { a.x = fmaxf(a.x, 0.0f); a.y = fmaxf(a.y, 0.0f); }
        #pragma unroll
        for (int t = 0; t < NT; ++t) {
            v2f b;
            b.x = B[(size_t)ka       * ncols + col[t]];
            b.y = B[(size_t)(ka + 1) * ncols + col[t]];
            if (GUARD) { b.x *= mask[t]; b.y *= mask[t]; }
            // 8 args: (neg_a, A, neg_b, B, c_mod, C, reuse_a, reuse_b)
            acc[t] = __builtin_amdgcn_wmma_f32_16x16x4_f32(
                         false, a, false, b, (short)0, acc[t], false, false);
        }
    }

    // Epilogue: dis[row]^2 is uniform across each 16-lane half.
    const int rbase = blockIdx.x * 16 + half * 8;
    float dsq[8];
    #pragma unroll
    for (int r = 0; r < 8; ++r) {
        int row = rbase + r;
        if (row >= nrows) row = nrows - 1;
        const float s = dis[row];
        dsq[r] = s * s;
    }

    #pragma unroll
    for (int t = 0; t < NT; ++t) {
        const int c0 = blockIdx.y * (16 * NT) + t * 16 + l;
        if (!GUARD || c0 < ncols) {
            const float bv = bias[c0];
            #pragma unroll
            for (int r = 0; r < 8; ++r) {
                const int row = rbase + r;
                if (row < nrows) {
                    const float v = acc[t][r];
                    XW   [(size_t)row * ncols + c0] = v;
                    Cinit[(size_t)row * ncols + c0] = v * dsq[r] + bv;
                }
            }
        }
    }
}

// ---------------------------------------------------------------------------
// out[dst[e], f] += feat[src[e], f] * dis[src]*dis[dst]
// One wave per edge; lanes stride the feature dimension (coalesced).
__global__ void k_edge_scatter(const int* __restrict__ src,
                               const int* __restrict__ dst,
                               const float* __restrict__ dis,
                               const float* __restrict__ feat,
                               float* __restrict__ out, int ncols, int E) {
    long long gid = (long long)blockIdx.x * blockDim.x + threadIdx.x;
    int e    = (int)(gid >> 5);
    int lane = (int)(gid & 31);
    if (e >= E) return;
    const int s = src[e];
    const int d = dst[e];
    const float norm = dis[s] * dis[d];
    const float* __restrict__ fp = feat + (size_t)s * ncols;
    float* __restrict__ op = out + (size_t)d * ncols;
    for (int f = lane; f < ncols; f += 32)
        atomicAdd(&op[f], fp[f] * norm);
}

// ---------------------------------------------------------------------------
extern "C" void kernel_launch(void* const* d_in, const int* in_sizes, int n_in,
                              void* d_out, int out_size, void* d_ws, size_t ws_size,
                              hipStream_t stream) {
    const float* x  = (const float*)d_in[0];
    const int*   ei = (const int*)  d_in[1];   // JAX default config -> int32
    const float* W1 = (const float*)d_in[2];
    const float* b1 = (const float*)d_in[3];
    const float* W2 = (const float*)d_in[4];
    const float* b2 = (const float*)d_in[5];
    float* out = (float*)d_out;

    const int N   = in_sizes[0] / NFEAT;       // 100000
    const int E   = in_sizes[1] / 2;           // 1600000
    const int CLS = in_sizes[5];               // 40  (b2 length)

    const int* src = ei;
    const int* dst = ei + E;

    // ---- workspace carve-up -------------------------------------------------
    char* ws = (char*)d_ws;
    size_t off = 0;
    float* dis = (float*)(ws + off); off += ((size_t)N * 4 + 255) & ~(size_t)255;
    float* xw1 = (float*)(ws + off); off += (size_t)N * NFEAT * 4;
    float* h   = (float*)(ws + off); off += (size_t)N * NFEAT * 4;
    float* hw2 = (float*)(ws + off); off += (size_t)N * CLS * 4;
    (void)ws_size; (void)n_in; (void)out_size;

    const int TPB = 256;

    // ---- normalization coefficients ----------------------------------------
    k_init_deg <<<(N + TPB - 1) / TPB, TPB, 0, stream>>>(dis, N);
    k_deg_accum<<<(E + TPB - 1) / TPB, TPB, 0, stream>>>(dst, dis, E);
    k_rsqrt    <<<(N + TPB - 1) / TPB, TPB, 0, stream>>>(dis, N);

    // ---- layer 1: xw1 = x @ W1 ; h = xw1*dis^2 + b1  (fused, fp32 WMMA) ----
    {
        dim3 grid((N + 15) / 16, NFEAT / 64);  // 6250 x 2; 2*4*16 == 128 cols exact
        k_wmma_gemm_fused<4, false, false><<<grid, 32, 0, stream>>>(
            x, W1, dis, b1, xw1, h, N, NFEAT);
    }
    // h += scatter_add(xw1[src] * norm)
    {
        long long threads = (long long)E * 32;
        k_edge_scatter<<<(unsigned)((threads + TPB - 1) / TPB), TPB, 0, stream>>>(
            src, dst, dis, xw1, h, NFEAT, E);
    }

    // ---- layer 2: hw2 = relu(h) @ W2 ; out = hw2*dis^2 + b2  (fused) -------
    {
        dim3 grid((N + 15) / 16, 1);           // 3 tiles cover 48 >= 40 cols (guarded)
        k_wmma_gemm_fused<3, true, true><<<grid, 32, 0, stream>>>(
            h, W2, dis, b2, hw2, out, N, CLS);
    }
    // out += scatter_add(hw2[src] * norm)
    {
        long long threads = (long long)E * 32;
        k_edge_scatter<<<(unsigned)((threads + TPB - 1) / TPB), TPB, 0, stream>>>(
            src, dst, dis, hw2, out, CLS, E);
    }
}